// GCN_89996744720764
// MI455X (gfx1250) — compile-verified
//
#include <hip/hip_runtime.h>

#define N_NODES 50000
#define N_EDGES 800000

typedef __attribute__((ext_vector_type(2))) float v2f;
typedef __attribute__((ext_vector_type(8))) float v8f;

// ---------------------------------------------------------------------------
// Degree accumulation: out_deg over src, in_deg over dst (fp32 atomics).
// ---------------------------------------------------------------------------
__global__ void gcn_deg_kernel(const int* __restrict__ src,
                               const int* __restrict__ dst,
                               float* __restrict__ odeg,
                               float* __restrict__ ideg, int E) {
  int i = blockIdx.x * blockDim.x + threadIdx.x;
  if (i < E) {
    atomicAdd(&odeg[src[i]], 1.0f);
    atomicAdd(&ideg[dst[i]], 1.0f);
  }
}

// deg -> 1/sqrt(max(deg,1)) in place, both arrays at once
__global__ void gcn_norm_kernel(float* __restrict__ a, float* __restrict__ b,
                                int n) {
  int i = blockIdx.x * blockDim.x + threadIdx.x;
  if (i < n) {
    a[i] = rsqrtf(fmaxf(a[i], 1.0f));
    b[i] = rsqrtf(fmaxf(b[i], 1.0f));
  }
}

// ---------------------------------------------------------------------------
// m = (h * out_norm_row) @ W  via V_WMMA_F32_16X16X4_F32.
// One wave computes a 16(node) x 16(outfeat) tile; K advances by 4 per WMMA.
// A frag (16x4 f32): lanes 0-15 hold K={k,k+1}, lanes 16-31 hold K={k+2,k+3},
//   row M = lane&15.  B frag (4x16) mirrors with lanes = N columns.
// C/D (16x16 f32): VGPR v -> M = v + 8*(lane>=16), N = lane&15.
// Column mask is branchless so EXEC stays all-1s through the WMMA.
// ---------------------------------------------------------------------------
__global__ void gcn_gemm_wmma_kernel(const float* __restrict__ h,
                                     const float* __restrict__ onorm,
                                     const float* __restrict__ W,
                                     float* __restrict__ m,
                                     int d_in, int d_out) {
  const int lane = threadIdx.x;        // 0..31
  const int half = lane >> 4;          // 0 | 1
  const int l15  = lane & 15;
  const int row  = blockIdx.x * 16 + l15;    // node row for A frag
  const int col  = blockIdx.y * 16 + l15;    // out-feature col for B frag
  const float an = onorm[row];

  const int   cvalid = (col < d_out) ? 1 : 0;
  const int   ccol   = cvalid ? col : 0;
  const float cmask  = (float)cvalid;

  const float* __restrict__ hrow = h + (long long)row * d_in;

  v8f acc = {};
  for (int kk = 0; kk < d_in; kk += 4) {
    const int ka = kk + 2 * half;
    v2f a, b;
    a.x = hrow[ka]     * an;
    a.y = hrow[ka + 1] * an;
    b.x = W[(long long)ka       * d_out + ccol] * cmask;
    b.y = W[(long long)(ka + 1) * d_out + ccol] * cmask;
    acc = __builtin_amdgcn_wmma_f32_16x16x4_f32(
        /*neg_a=*/false, a, /*neg_b=*/false, b,
        /*c_mod=*/(short)0, acc, /*reuse_a=*/false, /*reuse_b=*/false);
  }

  if (cvalid) {
    const int base_row = blockIdx.x * 16 + half * 8;
#pragma unroll
    for (int v = 0; v < 8; ++v) {
      m[(long long)(base_row + v) * d_out + col] = acc[v];
    }
  }
}

// ---------------------------------------------------------------------------
// Vectorized scatter-add: one thread per (edge, 4-feature chunk).
//   agg[dst[e]*d + 4c .. +3] += m[src[e]*d + 4c .. +3] * (use_ef ? efet[e]:1)
// Gather is a single global_load_b128; indices amortized over 16B payload.
// ld4 = log2(d/4); all layer dims are multiples of 4.
// ---------------------------------------------------------------------------
__global__ void gcn_scatter4_kernel(const float4* __restrict__ m4,
                                    const int* __restrict__ src,
                                    const int* __restrict__ dst,
                                    const float* __restrict__ efet,
                                    int use_ef,
                                    float* __restrict__ agg,
                                    int E, int ld4) {
  const int idx = blockIdx.x * blockDim.x + threadIdx.x;
  const int total = E << ld4;
  if (idx >= total) return;
  const int e  = idx >> ld4;
  const int c  = idx & ((1 << ld4) - 1);
  const int q  = 1 << ld4;              // d / 4
  float4 v = m4[src[e] * q + c];
  if (use_ef) {
    const float w = efet[e];
    v.x *= w; v.y *= w; v.z *= w; v.w *= w;
  }
  float* dp = agg + ((long long)dst[e] * q + c) * 4;
  atomicAdd(dp + 0, v.x);
  atomicAdd(dp + 1, v.y);
  atomicAdd(dp + 2, v.z);
  atomicAdd(dp + 3, v.w);
}

// out = (agg * in_norm_row + bias_col), optional ReLU
__global__ void gcn_post_kernel(const float* __restrict__ agg,
                                const float* __restrict__ inorm,
                                const float* __restrict__ bias,
                                float* __restrict__ out,
                                int n, int ld, int do_relu) {
  const int idx = blockIdx.x * blockDim.x + threadIdx.x;
  if (idx >= n) return;
  const int row = idx >> ld;
  const int f   = idx & ((1 << ld) - 1);
  float v = agg[idx] * inorm[row] + bias[f];
  out[idx] = do_relu ? fmaxf(v, 0.0f) : v;
}

__global__ void gcn_relu_kernel(const float* __restrict__ in,
                                float* __restrict__ out, int n) {
  const int idx = blockIdx.x * blockDim.x + threadIdx.x;
  if (idx < n) out[idx] = fmaxf(in[idx], 0.0f);
}

// ---------------------------------------------------------------------------
extern "C" void kernel_launch(void* const* d_in, const int* in_sizes, int n_in,
                              void* d_out, int out_size, void* d_ws,
                              size_t ws_size, hipStream_t stream) {
  (void)in_sizes; (void)n_in; (void)out_size; (void)ws_size;

  const float* x    = (const float*)d_in[0];
  const float* efet = (const float*)d_in[1];
  const int*   src  = (const int*)d_in[2];
  const int*   dst  = (const int*)d_in[3];
  const float* W[5] = {(const float*)d_in[4], (const float*)d_in[6],
                       (const float*)d_in[8], (const float*)d_in[10],
                       (const float*)d_in[12]};
  const float* B[5] = {(const float*)d_in[5], (const float*)d_in[7],
                       (const float*)d_in[9], (const float*)d_in[11],
                       (const float*)d_in[13]};

  const int dims[6] = {16, 256, 128, 64, 32, 4};
  const int lds[5]  = {8, 7, 6, 5, 2};   // log2(d_out) per layer
  const int N = N_NODES, E = N_EDGES;

  // workspace layout (fp32): onorm[N] | inorm[N] | bufA[N*256] | bufB[N*256]
  float* wsf   = (float*)d_ws;
  float* onorm = wsf;
  float* inorm = wsf + N;
  float* bufA  = wsf + 2 * (size_t)N;
  float* bufB  = bufA + (size_t)N * 256;

  // degrees -> norms
  hipMemsetAsync(onorm, 0, 2 * (size_t)N * sizeof(float), stream);
  gcn_deg_kernel<<<(E + 255) / 256, 256, 0, stream>>>(src, dst, onorm, inorm, E);
  gcn_norm_kernel<<<(N + 255) / 256, 256, 0, stream>>>(onorm, inorm, N);

  const float* h  = x;     // current layer input
  float* cur = bufA;       // gemm target / working buffer
  float* oth = bufB;       // scatter target

  for (int L = 0; L < 5; ++L) {
    const int din = dims[L], dout = dims[L + 1];
    const int ld  = lds[L];
    const int ld4 = ld - 2;                 // log2(dout/4)
    const int ntiles = (dout + 15) / 16;
    const int nelem  = N * dout;
    const int etot4  = E << ld4;            // (edge, float4-chunk) pairs

    // m = (h * onorm) @ W[L]   -> cur
    dim3 g(N / 16, ntiles);
    gcn_gemm_wmma_kernel<<<g, 32, 0, stream>>>(h, onorm, W[L], cur, din, dout);

    // agg = segment_sum(m[src], dst)  -> oth
    hipMemsetAsync(oth, 0, (size_t)nelem * sizeof(float), stream);
    gcn_scatter4_kernel<<<(etot4 + 255) / 256, 256, 0, stream>>>(
        (const float4*)cur, src, dst, efet, /*use_ef=*/0, oth, E, ld4);

    if (L == 4) {
      // final layer: h5 = agg*inorm + b  (no relu) straight into d_out
      gcn_post_kernel<<<(nelem + 255) / 256, 256, 0, stream>>>(
          oth, inorm, B[L], (float*)d_out, nelem, ld, /*relu=*/0);
      break;
    }

    // t = relu(agg*inorm + b)  -> cur (m no longer needed)
    gcn_post_kernel<<<(nelem + 255) / 256, 256, 0, stream>>>(
        oth, inorm, B[L], cur, nelem, ld, /*relu=*/1);

    if (L < 3) {
      // h_next = relu(segment_sum(t[src]*efet, dst))
      hipMemsetAsync(oth, 0, (size_t)nelem * sizeof(float), stream);
      gcn_scatter4_kernel<<<(etot4 + 255) / 256, 256, 0, stream>>>(
          (const float4*)cur, src, dst, efet, /*use_ef=*/1, oth, E, ld4);
      gcn_relu_kernel<<<(nelem + 255) / 256, 256, 0, stream>>>(oth, cur, nelem);
    }

    h = cur;                              // layer output lives in cur
    float* t = cur; cur = oth; oth = t;   // ping-pong for next layer
  }
}